// MultiLoraEmbedding_34007551050177
// MI455X (gfx1250) — compile-verified
//
#include <hip/hip_runtime.h>

// MultiLoraEmbedding: out[b,s,:] = weight[adapter_ids[b], x[b,s], :]
// B=32, S=2048, D=128, VOCAB=32000, MAX_LORAS=16.
// Pure gather, 0 FLOPs -> HBM-bandwidth bound (~64 MiB total traffic,
// ~2.9 us floor at 23.3 TB/s). One wave32 per token row: 32 lanes x 16 B
// = 512 B = one row. 4 tokens per wave.
//
// CDNA5 path: gather rows HBM -> LDS via GLOBAL_LOAD_ASYNC_TO_LDS_B128
// (ASYNCcnt), one s_wait_asynccnt 0, then stream LDS -> HBM via
// GLOBAL_STORE_ASYNC_FROM_LDS_B128. No VGPR staging; LDS bandwidth >> HBM
// so the roofline is unchanged. Per wave: 1 B128 index load + 1 B32
// adapter load + 4 async loads + 4 async stores.

#define EMBED_DIM       128
#define VOCAB           32000u
#define SEQ_LOG2        11        // SEQ = 2048
#define TOK_PER_WAVE    4
#define WAVES_PER_BLOCK 8         // 256 threads / wave32

typedef float v4f __attribute__((ext_vector_type(4)));
// Builtin signature (from clang diagnostic): pointers to 16-byte int
// vectors in global (AS1) and LDS (AS3) address spaces, then 2 immediates.
typedef int v4i __attribute__((vector_size(16)));
typedef __attribute__((address_space(1))) v4i glb_v4i;
typedef __attribute__((address_space(3))) v4i lds_v4i;

#if __has_builtin(__builtin_amdgcn_global_load_async_to_lds_b128) && \
    __has_builtin(__builtin_amdgcn_global_store_async_from_lds_b128) && \
    __has_builtin(__builtin_amdgcn_s_wait_asynccnt)
#define HAVE_ASYNC_LDS 1
#else
#define HAVE_ASYNC_LDS 0
#endif

__global__ __launch_bounds__(256) void multi_lora_embed_kernel(
    const int*  __restrict__ x,            // [B*S] token ids
    const int*  __restrict__ adapter_ids,  // [B]
    const float* __restrict__ weight,      // [MAX_LORAS, VOCAB, EMBED_DIM]
    float* __restrict__ out,               // [B*S, EMBED_DIM]
    int n_tokens)
{
#if HAVE_ASYNC_LDS
    __shared__ alignas(16) float lds_buf[WAVES_PER_BLOCK * TOK_PER_WAVE * EMBED_DIM]; // 16 KB
#endif
    const int lane  = threadIdx.x & 31;
    const int wslot = threadIdx.x >> 5;
    const int wave  = blockIdx.x * WAVES_PER_BLOCK + wslot;
    const int t0    = wave * TOK_PER_WAVE;

    if (t0 + TOK_PER_WAVE <= n_tokens) {
        // ---- fast path ----
        // 4 consecutive tokens never straddle a batch (4 | 2048): one
        // adapter load per wave; token ids as a single aligned B128 load.
        const unsigned aid_base =
            (unsigned)adapter_ids[t0 >> SEQ_LOG2] * VOCAB;
        const v4i w4 = *(const v4i*)(x + t0);

        // Row index fits in 32 bits: aid*VOCAB + word <= 512000.
        unsigned row[TOK_PER_WAVE];
#pragma unroll
        for (int i = 0; i < TOK_PER_WAVE; ++i)
            row[i] = aid_base + (unsigned)w4[i];

#if HAVE_ASYNC_LDS
        float* mybuf = lds_buf + wslot * (TOK_PER_WAVE * EMBED_DIM);
        // Gather: 4 async B128 transfers per lane, HBM -> LDS (ASYNCcnt).
#pragma unroll
        for (int i = 0; i < TOK_PER_WAVE; ++i) {
            const float* gsrc = weight + (size_t)row[i] * EMBED_DIM + lane * 4;
            float*       ldst = mybuf + i * EMBED_DIM + lane * 4;
            __builtin_amdgcn_global_load_async_to_lds_b128(
                (glb_v4i*)gsrc, (lds_v4i*)ldst, 0, 0);
        }
        __builtin_amdgcn_s_wait_asynccnt(0);
        // Stream-out: contiguous tile, LDS -> HBM (ASYNCcnt).
#pragma unroll
        for (int i = 0; i < TOK_PER_WAVE; ++i) {
            float* gdst = out + (size_t)(t0 + i) * EMBED_DIM + lane * 4;
            float* lsrc = mybuf + i * EMBED_DIM + lane * 4;
            __builtin_amdgcn_global_store_async_from_lds_b128(
                (glb_v4i*)gdst, (lds_v4i*)lsrc, 0, 0);
        }
        // S_ENDPGM performs an implicit wait-idle; async stores drain there.
#else
        v4f v[TOK_PER_WAVE];
#pragma unroll
        for (int i = 0; i < TOK_PER_WAVE; ++i) {
            const v4f* src = (const v4f*)(weight + (size_t)row[i] * EMBED_DIM);
            v[i] = src[lane];
        }
#pragma unroll
        for (int i = 0; i < TOK_PER_WAVE; ++i) {
            v4f* dst = (v4f*)(out + (size_t)(t0 + i) * EMBED_DIM);
            __builtin_nontemporal_store(v[i], dst + lane);
        }
#endif
    } else if (t0 < n_tokens) {
        // ---- tail path (never taken for the reference sizes) ----
        const unsigned aid_base =
            (unsigned)adapter_ids[t0 >> SEQ_LOG2] * VOCAB;
        for (int i = 0; i < TOK_PER_WAVE; ++i) {
            const int t = t0 + i;
            if (t < n_tokens) {
                const unsigned r = aid_base + (unsigned)x[t];
                const v4f* src = (const v4f*)(weight + (size_t)r * EMBED_DIM);
                v4f* dst = (v4f*)(out + (size_t)t * EMBED_DIM);
                __builtin_nontemporal_store(src[lane], dst + lane);
            }
        }
    }
}

extern "C" void kernel_launch(void* const* d_in, const int* in_sizes, int n_in,
                              void* d_out, int out_size, void* d_ws, size_t ws_size,
                              hipStream_t stream)
{
    const int*   x           = (const int*)  d_in[0];  // [B*S]
    const int*   adapter_ids = (const int*)  d_in[1];  // [B]
    const float* weight      = (const float*)d_in[2];  // [L, VOCAB, D]
    float*       out         = (float*)      d_out;    // [B*S, D]

    const int n_tokens = out_size / EMBED_DIM;         // 65536
    const int n_waves  = (n_tokens + TOK_PER_WAVE - 1) / TOK_PER_WAVE;
    const int n_blocks = (n_waves + WAVES_PER_BLOCK - 1) / WAVES_PER_BLOCK;

    multi_lora_embed_kernel<<<n_blocks, 256, 0, stream>>>(
        x, adapter_ids, weight, out, n_tokens);
}